// myNLLLoss_7962869367387
// MI455X (gfx1250) — compile-verified
//
#include <hip/hip_runtime.h>

// NLL loss with ignore_index=0:
//   loss = -sum_{i: target[i] != 0} output[i, target[i]]
// N = 4096 rows, V = 50257 classes, f32 logits, int32 targets, scalar f32 out.
//
// Latency-bound masked gather-reduce: one workgroup (32 wave32 waves) on one
// WGP, 4 rows per thread, branchless mask, wave-shuffle tree + LDS combine,
// single deterministic store (no atomics, graph-capture safe).
//
// All flat indices fit in 32 bits (4096*50257 = 205,852,672 < 2^31), so index
// math is kept in unsigned 32-bit: the compiler emits GVS addressing
// (SGPR base + 32-bit VGPR offset, scale_offset) with no 64-bit carry chains.

#ifndef NLL_N
#define NLL_N 4096
#endif
#ifndef NLL_V
#define NLL_V 50257
#endif

#define BLOCK 1024
#define ROWS_PER_THREAD (NLL_N / BLOCK)   // 4
#define WAVES (BLOCK / 32)                // 32

__global__ __launch_bounds__(BLOCK) void
myNLLLoss_7962869367387_kernel(const float* __restrict__ logits,
                               const int* __restrict__ target,
                               float* __restrict__ loss_out) {
    const unsigned tid = threadIdx.x;

    // --- Phase 1: issue all target loads (independent, coalesced) ---
    int t[ROWS_PER_THREAD];
#pragma unroll
    for (int k = 0; k < ROWS_PER_THREAD; ++k) {
        t[k] = target[tid + k * BLOCK];
    }

    // --- Phase 2: dependent gathers, 32-bit index math, issued back-to-back ---
    float acc = 0.0f;
#pragma unroll
    for (int k = 0; k < ROWS_PER_THREAD; ++k) {
        const unsigned row = tid + (unsigned)(k * BLOCK);
        // idx < 4096*50257 < 2^31: pure u32 mad, no 64-bit extend on the
        // critical path. Always in-bounds even when t[k]==0.
        const unsigned idx = row * (unsigned)NLL_V + (unsigned)t[k];
        const float v = logits[idx];
        acc += (t[k] != 0) ? v : 0.0f;   // branchless mask (v_cndmask)
    }

    // --- Phase 3: wave32 tree reduction (lane permutes, no memory traffic) ---
#pragma unroll
    for (int off = 16; off > 0; off >>= 1) {
        acc += __shfl_down(acc, off, 32);
    }

    // --- Phase 4: cross-wave combine through LDS ---
    __shared__ float wave_sum[WAVES];
    const unsigned lane = tid & 31u;
    const unsigned wave = tid >> 5;
    if (lane == 0) wave_sum[wave] = acc;
    __syncthreads();

    if (wave == 0) {
        float s = wave_sum[lane];   // WAVES == 32, one value per lane
#pragma unroll
        for (int off = 16; off > 0; off >>= 1) {
            s += __shfl_down(s, off, 32);
        }
        if (lane == 0) {
            loss_out[0] = -s;       // plain store: deterministic, overwrites poison
        }
    }
}

extern "C" void kernel_launch(void* const* d_in, const int* in_sizes, int n_in,
                              void* d_out, int out_size, void* d_ws, size_t ws_size,
                              hipStream_t stream) {
    const float* logits = (const float*)d_in[0];  // [4096, 50257] f32
    const int*   target = (const int*)d_in[1];    // [4096] int
    float*       loss   = (float*)d_out;          // [1] f32

    (void)in_sizes; (void)n_in; (void)out_size; (void)d_ws; (void)ws_size;

    myNLLLoss_7962869367387_kernel<<<1, BLOCK, 0, stream>>>(logits, target, loss);
}